// _PyHybridBlockRef_13219909337137
// MI455X (gfx1250) — compile-verified
//
#include <hip/hip_runtime.h>
#include <math.h>

// Problem constants (fixed by the reference)
#define BB   8
#define TT   2048
#define KIN  512
#define DD   1024
#define MM   (BB * TT)          // 16384 flattened rows
#define BTD  ((size_t)MM * DD)  // 16,777,216 elements per [B,T,D] tensor
#define LN_EPS 1e-5f

typedef __attribute__((ext_vector_type(2))) float v2f;
typedef __attribute__((ext_vector_type(8))) float v8f;

// ---------------------------------------------------------------------------
// Kernel 1: fused dual projection + pointwise epilogue.
//   yd = x@Wd^T + bd ; yb = x@Wb^T + bb
//   delta = softplus(yd) ; a = exp(-delta*exp(A_log)) ; bstep = delta*yb
// Each wave owns one 16x16 output tile and accumulates K=512 via 128
// V_WMMA_F32_16X16X4_F32 per weight matrix (A-fragment shared between both).
// Block = 4 waves stacked along M (covers 64 rows x 16 cols).
// ---------------------------------------------------------------------------
__global__ __launch_bounds__(128)
void proj_kernel(const float* __restrict__ x,
                 const float* __restrict__ Wd, const float* __restrict__ bd,
                 const float* __restrict__ Wb, const float* __restrict__ bb,
                 const float* __restrict__ A_log,
                 float* __restrict__ a_out, float* __restrict__ b_out)
{
    const int lane = threadIdx.x & 31;
    const int wave = threadIdx.x >> 5;
    const int row0 = (blockIdx.x * 4 + wave) * 16;   // M tile base
    const int n0   = blockIdx.y * 16;                // N tile base

    const int ml = lane & 15;            // M (for A) / N (for B) within tile
    const int kb = (lane >> 4) << 1;     // K sub-offset: 0 or 2

    const float* __restrict__ xrow  = x  + (size_t)(row0 + ml) * KIN + kb;
    const float* __restrict__ wdrow = Wd + (size_t)(n0  + ml) * KIN + kb;
    const float* __restrict__ wbrow = Wb + (size_t)(n0  + ml) * KIN + kb;

    v8f accd = {};
    v8f accb = {};

#pragma unroll 4
    for (int k = 0; k < KIN; k += 4) {
        v2f af  = *(const v2f*)(xrow  + k);   // A 16x4 fragment (K=kb..kb+1)
        v2f bdf = *(const v2f*)(wdrow + k);   // B 4x16 fragment for Wd
        v2f bbf = *(const v2f*)(wbrow + k);   // B 4x16 fragment for Wb
        accd = __builtin_amdgcn_wmma_f32_16x16x4_f32(
            false, af, false, bdf, (short)0, accd, false, false);
        accb = __builtin_amdgcn_wmma_f32_16x16x4_f32(
            false, af, false, bbf, (short)0, accb, false, false);
    }

    // Per-lane column index (N) is fixed across all 8 accumulator regs.
    const int n = n0 + ml;
    const float bdv = bd[n];
    const float bbv = bb[n];
    const float eA  = __expf(A_log[n]);
    const int mbase = row0 + ((lane >> 4) << 3);  // +8 rows for upper half-wave

#pragma unroll
    for (int r = 0; r < 8; ++r) {
        const int m = mbase + r;
        const float yd = accd[r] + bdv;
        const float yb = accb[r] + bbv;
        // numerically stable softplus (matches jax.nn.softplus)
        const float delta = fmaxf(yd, 0.0f) + log1pf(__expf(-fabsf(yd)));
        const float a  = __expf(-delta * eA);
        const float bs = delta * yb;
        const size_t idx = (size_t)m * DD + n;
        a_out[idx] = a;
        b_out[idx] = bs;
    }
}

// ---------------------------------------------------------------------------
// Kernel 2: sequential scan over T. One thread per (b,d) channel, 8192 total.
// Unrolled by 4 so 8 independent global loads are in flight per thread to
// hide HBM latency (only 256 waves exist, so ILP must carry the streaming).
// Writes h_post (post-reset) to out[0..BTD) and spikes to out[BTD..2*BTD).
// ---------------------------------------------------------------------------
__global__ __launch_bounds__(256)
void scan_kernel(const float* __restrict__ A, const float* __restrict__ Bv,
                 const float* __restrict__ h0, const float* __restrict__ thr,
                 float* __restrict__ hpost, float* __restrict__ spk)
{
    const int idx = blockIdx.x * 256 + threadIdx.x;   // 0 .. B*D-1
    const int b = idx / DD;
    const int d = idx - b * DD;

    float h = h0[idx];
    const float th = thr[d];
    const size_t base = (size_t)b * TT * DD + d;

    for (int t = 0; t < TT; t += 4) {
        const size_t p = base + (size_t)t * DD;
        // batch the loads (independent) before the dependent FMA chain
        const float a0 = A[p];          const float c0 = Bv[p];
        const float a1 = A[p + DD];     const float c1 = Bv[p + DD];
        const float a2 = A[p + 2*DD];   const float c2 = Bv[p + 2*DD];
        const float a3 = A[p + 3*DD];   const float c3 = Bv[p + 3*DD];

        float s;
        h = fmaf(a0, h, c0); s = (h > th) ? 1.0f : 0.0f; h = (s > 0.0f) ? 0.0f : h;
        hpost[p]        = h; spk[p]        = s;
        h = fmaf(a1, h, c1); s = (h > th) ? 1.0f : 0.0f; h = (s > 0.0f) ? 0.0f : h;
        hpost[p + DD]   = h; spk[p + DD]   = s;
        h = fmaf(a2, h, c2); s = (h > th) ? 1.0f : 0.0f; h = (s > 0.0f) ? 0.0f : h;
        hpost[p + 2*DD] = h; spk[p + 2*DD] = s;
        h = fmaf(a3, h, c3); s = (h > th) ? 1.0f : 0.0f; h = (s > 0.0f) ? 0.0f : h;
        hpost[p + 3*DD] = h; spk[p + 3*DD] = s;
    }
}

// ---------------------------------------------------------------------------
// Kernel 3: in-place LayerNorm over D (=1024) per row. One block (256 thr,
// 8 waves) per row; each thread owns one float4. Wave-shuffle + LDS reduce.
// ---------------------------------------------------------------------------
__global__ __launch_bounds__(256)
void ln_kernel(float* __restrict__ h,
               const float* __restrict__ gamma, const float* __restrict__ beta)
{
    __shared__ float s_sum[8];
    __shared__ float s_sq[8];
    __shared__ float s_stats[2];  // mean, inv_std

    float* __restrict__ row = h + (size_t)blockIdx.x * DD;
    const int d4 = threadIdx.x * 4;

    const float4 v = *(const float4*)(row + d4);
    float s = v.x + v.y + v.z + v.w;
    float q = v.x*v.x + v.y*v.y + v.z*v.z + v.w*v.w;

    // wave32 reduction
#pragma unroll
    for (int off = 16; off > 0; off >>= 1) {
        s += __shfl_down(s, off);
        q += __shfl_down(q, off);
    }
    if ((threadIdx.x & 31) == 0) {
        s_sum[threadIdx.x >> 5] = s;
        s_sq [threadIdx.x >> 5] = q;
    }
    __syncthreads();
    if (threadIdx.x == 0) {
        float S = 0.0f, Q = 0.0f;
#pragma unroll
        for (int i = 0; i < 8; ++i) { S += s_sum[i]; Q += s_sq[i]; }
        const float mean = S * (1.0f / DD);
        const float var  = Q * (1.0f / DD) - mean * mean;
        s_stats[0] = mean;
        s_stats[1] = rsqrtf(var + LN_EPS);
    }
    __syncthreads();
    const float mean = s_stats[0];
    const float inv  = s_stats[1];

    const float4 g = *(const float4*)(gamma + d4);
    const float4 be = *(const float4*)(beta + d4);
    float4 r;
    r.x = (v.x - mean) * inv * g.x + be.x;
    r.y = (v.y - mean) * inv * g.y + be.y;
    r.z = (v.z - mean) * inv * g.z + be.z;
    r.w = (v.w - mean) * inv * g.w + be.w;
    *(float4*)(row + d4) = r;
}

// ---------------------------------------------------------------------------
extern "C" void kernel_launch(void* const* d_in, const int* in_sizes, int n_in,
                              void* d_out, int out_size, void* d_ws, size_t ws_size,
                              hipStream_t stream)
{
    const float* x         = (const float*)d_in[0];
    const float* Wd        = (const float*)d_in[1];
    const float* bd        = (const float*)d_in[2];
    const float* Wb        = (const float*)d_in[3];
    const float* bb        = (const float*)d_in[4];
    const float* A_log     = (const float*)d_in[5];
    const float* threshold = (const float*)d_in[6];
    const float* gamma     = (const float*)d_in[7];
    const float* beta      = (const float*)d_in[8];
    const float* h0        = (const float*)d_in[9];

    float* out    = (float*)d_out;        // [B,T,D] normalized h
    float* spikes = out + BTD;            // [B,T,D] spikes

    // workspace: A_step (64MB) | b_step (64MB)
    float* a_ws = (float*)d_ws;
    float* b_ws = a_ws + BTD;

    // 1) dual-GEMM + pointwise -> A_step, b_step
    dim3 g1(MM / 64, DD / 16);            // 256 x 64 blocks, 4 waves each
    proj_kernel<<<g1, 128, 0, stream>>>(x, Wd, bd, Wb, bb, A_log, a_ws, b_ws);

    // 2) sequential spiking scan -> h_post (into out), spikes
    scan_kernel<<<(BB * DD) / 256, 256, 0, stream>>>(a_ws, b_ws, h0, threshold,
                                                     out, spikes);

    // 3) in-place LayerNorm over D
    ln_kernel<<<MM, 256, 0, stream>>>(out, gamma, beta);
}